// GCN_66013647339805
// MI455X (gfx1250) — compile-verified
//
#include <hip/hip_runtime.h>
#include <hip/hip_bf16.h>

#define DLAT 64
#define DFEAT 128
#define SLOPE 0.01f
#define EPSN 1e-12f

typedef __attribute__((ext_vector_type(2))) float v2f;
typedef __attribute__((ext_vector_type(8))) float v8f;

__device__ __forceinline__ float leaky(float v) { return v >= 0.0f ? v : SLOPE * v; }

// D = A(16x4) * B(4x16) + C, fp32, wave32.
// A layout (ISA 7.12.2): lanes 0-15 -> M=lane, vgpr{0,1}=K{0,1}; lanes 16-31 -> M=lane-16, K{2,3}.
// B layout (row-striped like C/D): lanes 0-15 -> N=lane, vgpr{0,1}=K{0,1}; lanes 16-31 -> K{2,3}.
__device__ __forceinline__ v8f wmma_f32(v2f a, v2f b, v8f c) {
    return __builtin_amdgcn_wmma_f32_16x16x4_f32(false, a, false, b, (short)0, c,
                                                 false, false);
}

// ---------------------------------------------------------------------------
// Kernel 1: x = L2normalize(features @ Wmlp + b);  y = x @ Wc1   (fused)
// 256 threads = 8 waves = 2 row-tiles x 4 col-tiles -> 32x64 output per block.
// ---------------------------------------------------------------------------
__global__ __launch_bounds__(256) void mlp_norm_conv_kernel(
    const float* __restrict__ feat, const float* __restrict__ Wmlp,
    const float* __restrict__ bmlp, const float* __restrict__ Wc,
    float* __restrict__ xout, float* __restrict__ yout, int nnode)
{
    __shared__ float Wm_s[DFEAT * DLAT];   // 32 KB
    __shared__ float Wc_s[DLAT * DLAT];    // 16 KB
    __shared__ float xt[32 * 65];          // padded tile (stride 65: conflict-free)
    __shared__ float scl[32];

    const int tid = threadIdx.x;
    for (int i = tid; i < DFEAT * DLAT; i += 256) Wm_s[i] = Wmlp[i];
    for (int i = tid; i < DLAT * DLAT; i += 256) Wc_s[i] = Wc[i];
    __syncthreads();

    const int lane    = tid & 31;
    const int wave    = tid >> 5;
    const int rowtile = wave >> 2;           // 0..1
    const int coltile = wave & 3;            // 0..3
    const int mn      = lane & 15;
    const int kh      = (lane >> 4) << 1;    // 0 or 2
    const int col     = coltile * 16 + mn;
    const int blk0    = blockIdx.x * 32;

    int arow = blk0 + rowtile * 16 + mn;     // clamp (not mask) so EXEC stays all-1
    if (arow > nnode - 1) arow = nnode - 1;
    const float* ap = feat + (size_t)arow * DFEAT + kh;

    v8f acc = {};
    #pragma unroll
    for (int kk = 0; kk < DFEAT; kk += 4) {
        v2f a = *(const v2f*)(ap + kk);
        v2f b;
        b.x = Wm_s[(kk + kh) * DLAT + col];
        b.y = Wm_s[(kk + kh + 1) * DLAT + col];
        acc = wmma_f32(a, b, acc);
    }

    const float bias  = bmlp[col];
    const int   rbase = rowtile * 16 + ((lane >> 4) << 3);
    #pragma unroll
    for (int i = 0; i < 8; ++i)
        xt[(rbase + i) * 65 + col] = acc[i] + bias;
    __syncthreads();

    if (tid < 32) {                          // row-wise L2 norm of 32x64 tile
        float s = 0.f;
        #pragma unroll
        for (int j = 0; j < DLAT; ++j) { float v = xt[tid * 65 + j]; s += v * v; }
        scl[tid] = 1.0f / fmaxf(sqrtf(s), EPSN);
    }
    __syncthreads();

    for (int e = tid; e < 32 * DLAT; e += 256) {
        int r = e >> 6, c = e & 63;
        float v = xt[r * 65 + c] * scl[r];
        xt[r * 65 + c] = v;
        int g = blk0 + r;
        if (g < nnode) xout[(size_t)g * DLAT + c] = v;
    }
    __syncthreads();

    // y = x @ Wc1 (A from LDS tile)
    v8f acc2 = {};
    const int rl = rowtile * 16 + mn;
    #pragma unroll
    for (int kk = 0; kk < DLAT; kk += 4) {
        v2f a;
        a.x = xt[rl * 65 + kk + kh];
        a.y = xt[rl * 65 + kk + kh + 1];
        v2f b;
        b.x = Wc_s[(kk + kh) * DLAT + col];
        b.y = Wc_s[(kk + kh + 1) * DLAT + col];
        acc2 = wmma_f32(a, b, acc2);
    }
    #pragma unroll
    for (int i = 0; i < 8; ++i) {
        int g = blk0 + rbase + i;
        if (g < nnode) yout[(size_t)g * DLAT + col] = acc2[i];
    }
}

// ---------------------------------------------------------------------------
// Node update: h=leaky(hraw); x' = leaky(h@Wg+bg + leaky(x@Wl+bl) + id)
// FUSE_NEXT: additionally y = x' @ Wc (next layer conv), x' staged in LDS so
// the in-place x write happens only after every wave finished reading x.
// ---------------------------------------------------------------------------
template <bool FUSE_NEXT>
__global__ __launch_bounds__(256) void update_kernel(
    const float* __restrict__ xin, const float* __restrict__ hraw,
    const float* __restrict__ idemb,
    const float* __restrict__ Wl, const float* __restrict__ bl,
    const float* __restrict__ Wg, const float* __restrict__ bg,
    const float* __restrict__ Wc, float* __restrict__ xout,
    float* __restrict__ yout, int nnode)
{
    __shared__ float Wl_s[DLAT * DLAT];
    __shared__ float Wg_s[DLAT * DLAT];
    __shared__ float Wc_s[DLAT * DLAT];
    __shared__ float xt[32 * 65];

    const int tid = threadIdx.x;
    for (int i = tid; i < DLAT * DLAT; i += 256) {
        Wl_s[i] = Wl[i];
        Wg_s[i] = Wg[i];
        if constexpr (FUSE_NEXT) Wc_s[i] = Wc[i];
    }
    __syncthreads();

    const int lane    = tid & 31;
    const int wave    = tid >> 5;
    const int rowtile = wave >> 2;
    const int coltile = wave & 3;
    const int mn      = lane & 15;
    const int kh      = (lane >> 4) << 1;
    const int col     = coltile * 16 + mn;
    const int blk0    = blockIdx.x * 32;

    int arow = blk0 + rowtile * 16 + mn;
    if (arow > nnode - 1) arow = nnode - 1;
    const float* xp = xin + (size_t)arow * DLAT + kh;
    const float* hp = hraw + (size_t)arow * DLAT + kh;

    v8f c1 = {}, c2 = {};
    #pragma unroll
    for (int kk = 0; kk < DLAT; kk += 4) {
        v2f ax = *(const v2f*)(xp + kk);
        v2f hv = *(const v2f*)(hp + kk);
        v2f ah; ah.x = leaky(hv.x); ah.y = leaky(hv.y);   // h = leaky(conv) fused into A-frag
        v2f b1;
        b1.x = Wl_s[(kk + kh) * DLAT + col];
        b1.y = Wl_s[(kk + kh + 1) * DLAT + col];
        v2f b2;
        b2.x = Wg_s[(kk + kh) * DLAT + col];
        b2.y = Wg_s[(kk + kh + 1) * DLAT + col];
        c1 = wmma_f32(ax, b1, c1);
        c2 = wmma_f32(ah, b2, c2);
    }

    const float blv = bl[col], bgv = bg[col];
    const int rbase = rowtile * 16 + ((lane >> 4) << 3);
    #pragma unroll
    for (int i = 0; i < 8; ++i) {
        int rloc = rbase + i;
        int g    = blk0 + rloc;
        int gc   = g > nnode - 1 ? nnode - 1 : g;
        float idv  = idemb[(size_t)gc * DLAT + col];
        float xhat = leaky(c1[i] + blv) + idv;
        float v    = leaky(c2[i] + bgv + xhat);
        if constexpr (FUSE_NEXT) {
            xt[rloc * 65 + col] = v;                 // stage; global write after barrier
        } else {
            if (g < nnode) xout[(size_t)g * DLAT + col] = v;
        }
    }

    if constexpr (FUSE_NEXT) {
        __syncthreads();   // all reads of xin done -> safe in-place write of xout==xin
        for (int e = tid; e < 32 * DLAT; e += 256) {
            int r = e >> 6, c = e & 63;
            int g = blk0 + r;
            if (g < nnode) xout[(size_t)g * DLAT + c] = xt[r * 65 + c];
        }
        v8f acc2 = {};
        const int rl = rowtile * 16 + mn;
        #pragma unroll
        for (int kk = 0; kk < DLAT; kk += 4) {
            v2f a;
            a.x = xt[rl * 65 + kk + kh];
            a.y = xt[rl * 65 + kk + kh + 1];
            v2f b;
            b.x = Wc_s[(kk + kh) * DLAT + col];
            b.y = Wc_s[(kk + kh + 1) * DLAT + col];
            acc2 = wmma_f32(a, b, acc2);
        }
        #pragma unroll
        for (int i = 0; i < 8; ++i) {
            int g = blk0 + rbase + i;
            if (g < nnode) yout[(size_t)g * DLAT + col] = acc2[i];
        }
    }
}

// ---------------------------------------------------------------------------
// h[dst] += y[src] over edges. One wave per edge, float2 per lane (64 floats).
// ---------------------------------------------------------------------------
__global__ __launch_bounds__(256) void scatter_kernel(
    const float* __restrict__ y, const int* __restrict__ src,
    const int* __restrict__ dst, float* __restrict__ h, int nedge)
{
    const int lane = threadIdx.x & 31;
    const int wid  = (blockIdx.x * blockDim.x + threadIdx.x) >> 5;
    const int nw   = (gridDim.x * blockDim.x) >> 5;
    for (int e = wid; e < nedge; e += nw) {
        const int s = src[e], d = dst[e];
        v2f v = *(const v2f*)(y + (size_t)s * DLAT + lane * 2);
        float* hp = h + (size_t)d * DLAT + lane * 2;
        unsafeAtomicAdd(hp, v.x);       // -> global_atomic_add_f32 (no CAS loop)
        unsafeAtomicAdd(hp + 1, v.y);
    }
}

__global__ __launch_bounds__(256) void zero_kernel(float4* __restrict__ p, size_t n4)
{
    size_t i      = (size_t)blockIdx.x * blockDim.x + threadIdx.x;
    size_t stride = (size_t)gridDim.x * blockDim.x;
    float4 z = {0.f, 0.f, 0.f, 0.f};
    for (; i < n4; i += stride) p[i] = z;
}

extern "C" void kernel_launch(void* const* d_in, const int* in_sizes, int n_in,
                              void* d_out, int out_size, void* d_ws, size_t ws_size,
                              hipStream_t stream)
{
    const float* features = (const float*)d_in[0];
    const float* id_emb   = (const float*)d_in[1];
    const int*   edge     = (const int*)d_in[2];
    const float* Wmlp     = (const float*)d_in[3];
    const float* bmlp     = (const float*)d_in[4];
    const float* Wc1      = (const float*)d_in[5];
    const float* Wl1      = (const float*)d_in[6];
    const float* bl1      = (const float*)d_in[7];
    const float* Wg1      = (const float*)d_in[8];
    const float* bg1      = (const float*)d_in[9];
    const float* Wc2      = (const float*)d_in[10];
    const float* Wl2      = (const float*)d_in[11];
    const float* bl2      = (const float*)d_in[12];
    const float* Wg2      = (const float*)d_in[13];
    const float* bg2      = (const float*)d_in[14];

    const int nnode = in_sizes[0] / DFEAT;
    const int nedge = in_sizes[2] / 2;
    const int* src  = edge;
    const int* dst  = edge + nedge;

    float* x = (float*)d_ws;                     // [N,64]
    float* h = x + (size_t)nnode * DLAT;         // [N,64]
    float* y = h + (size_t)nnode * DLAT;         // [N,64]

    const int    nblk = (nnode + 31) / 32;
    const size_t hn4  = ((size_t)nnode * DLAT) / 4;

    // x = normalize(feat@Wmlp+b); y = x@Wc1
    mlp_norm_conv_kernel<<<nblk, 256, 0, stream>>>(features, Wmlp, bmlp, Wc1, x, y, nnode);
    // layer 1 conv scatter
    zero_kernel<<<2048, 256, 0, stream>>>((float4*)h, hn4);
    scatter_kernel<<<2048, 256, 0, stream>>>(y, src, dst, h, nedge);
    // layer 1 update (in-place x) + y = x@Wc2
    update_kernel<true><<<nblk, 256, 0, stream>>>(x, h, id_emb, Wl1, bl1, Wg1, bg1,
                                                  Wc2, x, y, nnode);
    // layer 2 conv scatter
    zero_kernel<<<2048, 256, 0, stream>>>((float4*)h, hn4);
    scatter_kernel<<<2048, 256, 0, stream>>>(y, src, dst, h, nedge);
    // layer 2 update -> d_out
    update_kernel<false><<<nblk, 256, 0, stream>>>(x, h, id_emb, Wl2, bl2, Wg2, bg2,
                                                   nullptr, (float*)d_out, nullptr, nnode);
}